// LightGCN_72232759984224
// MI455X (gfx1250) — compile-verified
//
#include <hip/hip_runtime.h>
#include <hip/hip_bf16.h>

#define NUM_USERS 400000
#define NUM_ITEMS 200000
#define NTOT      (NUM_USERS + NUM_ITEMS)   // 600000
#define DDIM      64
#define NNZ_      4000000
#define BATCH_    16384
#define NTILES    (NTOT / 16)               // 37500

typedef __attribute__((ext_vector_type(2))) float v2f;
typedef __attribute__((ext_vector_type(8))) float v8f;

// ---------------------------------------------------------------------------
// f0 = concat(user_emb, item_emb)  (float4 vectorized copy)
// ---------------------------------------------------------------------------
__global__ __launch_bounds__(256) void concat_f0(const float* __restrict__ ue,
                                                 const float* __restrict__ ie,
                                                 float* __restrict__ f0) {
    size_t i  = (size_t)blockIdx.x * blockDim.x + threadIdx.x;   // float4 index
    size_t nU = (size_t)NUM_USERS * DDIM / 4;
    size_t nT = (size_t)NTOT * DDIM / 4;
    if (i >= nT) return;
    float4 v = (i < nU) ? ((const float4*)ue)[i] : ((const float4*)ie)[i - nU];
    ((float4*)f0)[i] = v;
}

__global__ __launch_bounds__(256) void zero_f(float* __restrict__ p, long n4) {
    long i = (long)blockIdx.x * blockDim.x + threadIdx.x;
    if (i >= n4) return;
    ((float4*)p)[i] = make_float4(0.f, 0.f, 0.f, 0.f);
}

// ---------------------------------------------------------------------------
// y += scatter_row( val * x[col] )        (16 lanes per edge, float4 each)
// ---------------------------------------------------------------------------
__global__ __launch_bounds__(256) void spmm_scatter(const int*   __restrict__ row,
                                                    const int*   __restrict__ col,
                                                    const float* __restrict__ val,
                                                    const float* __restrict__ x,
                                                    float*       __restrict__ y) {
    int t  = blockIdx.x * 256 + threadIdx.x;
    int e  = t >> 4;
    int d0 = (t & 15) * 4;
    if (e >= NNZ_) return;
    int   r = row[e], c = col[e];
    float v = val[e];
    const float4 xv = *(const float4*)(x + (size_t)c * DDIM + d0);
    float* yp = y + (size_t)r * DDIM + d0;
    __hip_atomic_fetch_add(yp + 0, v * xv.x, __ATOMIC_RELAXED, __HIP_MEMORY_SCOPE_AGENT);
    __hip_atomic_fetch_add(yp + 1, v * xv.y, __ATOMIC_RELAXED, __HIP_MEMORY_SCOPE_AGENT);
    __hip_atomic_fetch_add(yp + 2, v * xv.z, __ATOMIC_RELAXED, __HIP_MEMORY_SCOPE_AGENT);
    __hip_atomic_fetch_add(yp + 3, v * xv.w, __ATOMIC_RELAXED, __HIP_MEMORY_SCOPE_AGENT);
}

// y += scatter_row( val * (lf[col] * f[col]) )   -- fuses inter = Lf*F
__global__ __launch_bounds__(256) void spmm_scatter_mul(const int*   __restrict__ row,
                                                        const int*   __restrict__ col,
                                                        const float* __restrict__ val,
                                                        const float* __restrict__ lf,
                                                        const float* __restrict__ f,
                                                        float*       __restrict__ y) {
    int t  = blockIdx.x * 256 + threadIdx.x;
    int e  = t >> 4;
    int d0 = (t & 15) * 4;
    if (e >= NNZ_) return;
    int   r = row[e], c = col[e];
    float v = val[e];
    size_t off = (size_t)c * DDIM + d0;
    const float4 a = *(const float4*)(lf + off);
    const float4 b = *(const float4*)(f + off);
    float* yp = y + (size_t)r * DDIM + d0;
    __hip_atomic_fetch_add(yp + 0, v * a.x * b.x, __ATOMIC_RELAXED, __HIP_MEMORY_SCOPE_AGENT);
    __hip_atomic_fetch_add(yp + 1, v * a.y * b.y, __ATOMIC_RELAXED, __HIP_MEMORY_SCOPE_AGENT);
    __hip_atomic_fetch_add(yp + 2, v * a.z * b.z, __ATOMIC_RELAXED, __HIP_MEMORY_SCOPE_AGENT);
    __hip_atomic_fetch_add(yp + 3, v * a.w * b.w, __ATOMIC_RELAXED, __HIP_MEMORY_SCOPE_AGENT);
}

// ---------------------------------------------------------------------------
// Fout = (Lf + Fin) @ W1^T + Linter @ W2^T + (b1 + b2)
// One wave32 per 16-row tile; V_WMMA_F32_16X16X4_F32.
//
// Weights staged in LDS as K-PAIRS (v2f elements), skewed stride 65:
//   sW[kp*65 + j] = { W[j, 2*kp], W[j, 2*kp+1] }       kp in [0,32), j in [0,64)
// B frag (4x16 f32, 2 VGPRs): lane n = lane&15, K half-select = lane>>4:
//   b = sW[(k0/2 + (lane>>4))*65 + j]   -> ONE aligned ds_load_b64, no shuffles.
// A frag (16x4 f32, 2 VGPRs): M = lane&15, VGPR0/1 hold K = koff, koff+1,
//   koff = (lane>=16) ? 2 : 0.
// C/D (16x16 f32, 8 VGPRs): row = v + 8*(lane>=16), col = j0 + (lane&15).
// ---------------------------------------------------------------------------
__global__ __launch_bounds__(256) void gnn_gemm(const float* __restrict__ Lf,
                                                const float* __restrict__ Fin,
                                                const float* __restrict__ Linter,
                                                const float* __restrict__ W1,
                                                const float* __restrict__ W2,
                                                const float* __restrict__ B1,
                                                const float* __restrict__ B2,
                                                float* __restrict__ Fout) {
    __shared__ v2f  sW1[32 * 65 + 64];          // kp*65 + j
    __shared__ v2f  sW2[32 * 65 + 64];
    __shared__ float sB[64];

    int tid = threadIdx.x;
    // 2048 float2 elements per matrix; 8 per thread
    for (int i = tid; i < 2048; i += 256) {
        int j  = i >> 5;                        // row (output feature)
        int kp = i & 31;                        // K pair index
        sW1[kp * 65 + j] = *(const v2f*)(W1 + (size_t)j * 64 + 2 * kp);
        sW2[kp * 65 + j] = *(const v2f*)(W2 + (size_t)j * 64 + 2 * kp);
    }
    if (tid < 64) sB[tid] = B1[tid] + B2[tid];
    __syncthreads();

    int wave = tid >> 5, lane = tid & 31;
    int tile = blockIdx.x * 8 + wave;
    if (tile >= NTILES) return;                 // wave-uniform: EXEC all-1s inside

    int row0  = tile * 16;
    int m     = lane & 15;
    int hi    = lane >> 4;                      // 0 lower half, 1 upper half
    int koff  = hi << 1;                        // 0 or 2
    int bbase = hi * 65 + m;                    // LDS v2f index base for B frags

    const float* lfp = Lf     + (size_t)(row0 + m) * DDIM + koff;
    const float* fip = Fin    + (size_t)(row0 + m) * DDIM + koff;
    const float* lip = Linter + (size_t)(row0 + m) * DDIM + koff;

    v8f acc[4];
    acc[0] = (v8f){}; acc[1] = (v8f){}; acc[2] = (v8f){}; acc[3] = (v8f){};

    #pragma unroll 4
    for (int k0 = 0; k0 < 64; k0 += 4) {
        v2f lf2 = *(const v2f*)(lfp + k0);
        v2f fi2 = *(const v2f*)(fip + k0);
        v2f li2 = *(const v2f*)(lip + k0);
        v2f a1; a1.x = lf2.x + fi2.x; a1.y = lf2.y + fi2.y;   // (Lf + F) frag
        v2f a2 = li2;                                          // Linter frag
        int kidx = (k0 >> 1) * 65 + bbase;                     // v2f index
        #pragma unroll
        for (int jt = 0; jt < 4; ++jt) {
            v2f b1 = sW1[kidx + jt * 16];
            v2f b2 = sW2[kidx + jt * 16];
            acc[jt] = __builtin_amdgcn_wmma_f32_16x16x4_f32(
                false, a1, false, b1, (short)0, acc[jt], false, false);
            acc[jt] = __builtin_amdgcn_wmma_f32_16x16x4_f32(
                false, a2, false, b2, (short)0, acc[jt], false, false);
        }
    }

    int rbase = row0 + (hi << 3);               // +8 for upper lane half
    #pragma unroll
    for (int jt = 0; jt < 4; ++jt) {
        float bv = sB[jt * 16 + m];
        #pragma unroll
        for (int v = 0; v < 8; ++v) {
            Fout[(size_t)(rbase + v) * DDIM + jt * 16 + m] = acc[jt][v] + bv;
        }
    }
}

// ---------------------------------------------------------------------------
// MLP head: gather [f0|f1|f2] rows for (uid, iid+NUM_USERS) -> [384] in LDS,
// fc1(384->64)+leaky, fc2(64->32)+leaky, fc3(32->1). One block per batch row.
// ---------------------------------------------------------------------------
__global__ __launch_bounds__(64) void mlp_head(const int*   __restrict__ uids,
                                               const int*   __restrict__ iids,
                                               const float* __restrict__ f0,
                                               const float* __restrict__ f1,
                                               const float* __restrict__ f2,
                                               const float* __restrict__ w1,
                                               const float* __restrict__ b1,
                                               const float* __restrict__ w2,
                                               const float* __restrict__ b2,
                                               const float* __restrict__ w3,
                                               const float* __restrict__ b3,
                                               float* __restrict__ out) {
    __shared__ float se[384];
    __shared__ float sh1[64];
    __shared__ float sh2[32];

    int b  = blockIdx.x;
    int t  = threadIdx.x;
    int u  = uids[b];
    int it = iids[b] + NUM_USERS;

    for (int idx = t; idx < 384; idx += 64) {
        int half = idx >= 192;
        int rr   = half ? it : u;
        int k    = idx - half * 192;
        int s    = k >> 6, d = k & 63;
        const float* src = (s == 0) ? f0 : ((s == 1) ? f1 : f2);
        se[idx] = src[(size_t)rr * DDIM + d];
    }
    __syncthreads();

    {
        float a = b1[t];
        const float* wr = w1 + t * 384;
        for (int k = 0; k < 384; ++k) a += se[k] * wr[k];
        sh1[t] = a > 0.f ? a : 0.01f * a;
    }
    __syncthreads();

    if (t < 32) {
        float a = b2[t];
        const float* wr = w2 + t * 64;
        for (int k = 0; k < 64; ++k) a += sh1[k] * wr[k];
        sh2[t] = a > 0.f ? a : 0.01f * a;
    }
    __syncthreads();

    if (t == 0) {
        float a = b3[0];
        for (int k = 0; k < 32; ++k) a += sh2[k] * w3[k];
        out[b] = a;
    }
}

// ---------------------------------------------------------------------------
extern "C" void kernel_launch(void* const* d_in, const int* in_sizes, int n_in,
                              void* d_out, int out_size, void* d_ws, size_t ws_size,
                              hipStream_t stream) {
    (void)in_sizes; (void)n_in; (void)out_size; (void)ws_size;

    const int*   uids  = (const int*)  d_in[0];
    const int*   iids  = (const int*)  d_in[1];
    const int*   erow  = (const int*)  d_in[2];
    const int*   ecol  = (const int*)  d_in[3];
    const float* evalv = (const float*)d_in[4];
    const float* uemb  = (const float*)d_in[5];
    const float* iemb  = (const float*)d_in[6];
    const float* gw1   = (const float*)d_in[7];
    const float* gb1   = (const float*)d_in[8];
    const float* gw2   = (const float*)d_in[9];
    const float* gb2   = (const float*)d_in[10];
    const float* fc1w  = (const float*)d_in[11];
    const float* fc1b  = (const float*)d_in[12];
    const float* fc2w  = (const float*)d_in[13];
    const float* fc2b  = (const float*)d_in[14];
    const float* fc3w  = (const float*)d_in[15];
    const float* fc3b  = (const float*)d_in[16];
    float* out = (float*)d_out;

    const size_t S = (size_t)NTOT * DDIM;       // elements per [N,64] buffer
    float* f0 = (float*)d_ws;
    float* f1 = f0 + S;
    float* f2 = f1 + S;
    float* Lf = f2 + S;
    float* Li = Lf + S;

    const long n4        = (long)(S / 4);
    const int  vecBlocks = (int)((n4 + 255) / 256);
    const int  spmmBlks  = NNZ_ / 16;           // 16 edges per 256-thread block
    const int  gemmBlks  = (NTILES + 7) / 8;

    concat_f0<<<vecBlocks, 256, 0, stream>>>(uemb, iemb, f0);

    const float* fin = f0;
    float* fouts[2] = {f1, f2};
    for (int l = 0; l < 2; ++l) {
        zero_f<<<vecBlocks, 256, 0, stream>>>(Lf, n4);
        spmm_scatter<<<spmmBlks, 256, 0, stream>>>(erow, ecol, evalv, fin, Lf);
        zero_f<<<vecBlocks, 256, 0, stream>>>(Li, n4);
        spmm_scatter_mul<<<spmmBlks, 256, 0, stream>>>(erow, ecol, evalv, Lf, fin, Li);
        gnn_gemm<<<gemmBlks, 256, 0, stream>>>(Lf, fin, Li,
                                               gw1 + (size_t)l * 4096,
                                               gw2 + (size_t)l * 4096,
                                               gb1 + (size_t)l * 64,
                                               gb2 + (size_t)l * 64,
                                               fouts[l]);
        fin = fouts[l];
    }

    mlp_head<<<BATCH_, 64, 0, stream>>>(uids, iids, f0, f1, f2,
                                        fc1w, fc1b, fc2w, fc2b, fc3w, fc3b, out);
}